// CategoricalActorCriticNet_HNet_CL_68066641707831
// MI455X (gfx1250) — compile-verified
//
#include <hip/hip_runtime.h>

#define S_DIM 1024
#define A_DIM 512
#define T_DIM 64
#define E_DIM 64
#define H_DIM 128
#define B_N   256
#define AW2_STRIDE (A_DIM * S_DIM)   // 524288 floats per aw2 row

typedef float v2f __attribute__((ext_vector_type(2)));
typedef float v8f __attribute__((ext_vector_type(8)));
typedef unsigned int v4u __attribute__((ext_vector_type(4)));
typedef int v4i __attribute__((ext_vector_type(4)));
typedef int v8i __attribute__((ext_vector_type(8)));

#if __has_builtin(__builtin_amdgcn_tensor_load_to_lds)
#define USE_TDM 1
#else
#define USE_TDM 0
#endif

// ---------------- obs [B,S] -> obsP pair-interleaved [S/2][B][2] ----------------
// obsP[(i>>1)*2*B + 2*b + (i&1)] = obs[b][i]  -> WMMA B operand {B[k][n],B[k+1][n]} contiguous
__global__ void k_transpose(const float* __restrict__ obs, float* __restrict__ obsP) {
  __shared__ float tile[32][33];
  const int i0 = blockIdx.x * 32;   // state index
  const int b0 = blockIdx.y * 32;   // batch index
  const int tx = threadIdx.x;       // 0..31
  const int ty = threadIdx.y;       // 0..7
  for (int r = ty; r < 32; r += 8)
    tile[r][tx] = obs[(b0 + r) * S_DIM + i0 + tx];
  __syncthreads();
  for (int r = ty; r < 32; r += 8) {
    const int i = i0 + r;
    obsP[(i >> 1) * (2 * B_N) + 2 * (b0 + tx) + (i & 1)] = tile[tx][r];
  }
}

// ---------------- task encoder: z = relu(relu(tl@we1+be1)@we2+be2) ----------------
__global__ void k_taskmlp(const float* __restrict__ tl,
                          const float* __restrict__ we1, const float* __restrict__ be1,
                          const float* __restrict__ we2, const float* __restrict__ be2,
                          float* __restrict__ z) {
  const int b = blockIdx.x;
  const int j = threadIdx.x;        // 64 threads
  __shared__ float tin[T_DIM];
  __shared__ float z1[E_DIM];
  tin[j] = tl[b * T_DIM + j];
  __syncthreads();
  float acc = be1[j];
  for (int k = 0; k < T_DIM; ++k) acc += tin[k] * we1[k * E_DIM + j];
  z1[j] = fmaxf(acc, 0.f);
  __syncthreads();
  acc = be2[j];
  for (int k = 0; k < E_DIM; ++k) acc += z1[k] * we2[k * E_DIM + j];
  z[b * E_DIM + j] = fmaxf(acc, 0.f);
}

// ---------------- fold aw2b bias: baF[o][b] = aw2b[o,:] . obs[b,:] ----------------
__global__ void k_pb(const float* __restrict__ aw2b, const float* __restrict__ obsP,
                     float* __restrict__ baF) {
  const int o = blockIdx.x;
  const int t = threadIdx.x;        // 256 threads = batch
  __shared__ float wb[S_DIM];
  for (int i = t; i < S_DIM; i += B_N) wb[i] = aw2b[o * S_DIM + i];
  __syncthreads();
  float acc = 0.f;
  for (int ip = 0; ip < S_DIM / 2; ++ip) {
    const float2 p = *(const float2*)&obsP[ip * (2 * B_N) + 2 * t];
    acc += wb[2 * ip] * p.x + wb[2 * ip + 1] * p.y;
  }
  baF[o * B_N + t] = acc;
}

// ------- hidden layers: h_aT, ba fold, full critic path v -------
__global__ void k_hidden(const float* __restrict__ z, const float* __restrict__ obs,
    const float* __restrict__ aw1, const float* __restrict__ aw1b,
    const float* __restrict__ ab1, const float* __restrict__ ab1b,
    const float* __restrict__ ab2, const float* __restrict__ ab2b,
    const float* __restrict__ cw1, const float* __restrict__ cw1b,
    const float* __restrict__ cw2, const float* __restrict__ cw2b,
    const float* __restrict__ cb1, const float* __restrict__ cb1b,
    const float* __restrict__ cb2, const float* __restrict__ cb2b,
    float* __restrict__ h_aT, float* __restrict__ baF, float* __restrict__ out_v) {
  const int b = blockIdx.x;
  const int t = threadIdx.x;        // 128 threads
  __shared__ float zl[E_DIM];
  __shared__ float hb[H_DIM];
  __shared__ float hc[H_DIM];
  __shared__ float hcb[H_DIM];
  __shared__ float red[H_DIM];
  if (t < E_DIM) zl[t] = z[b * E_DIM + t];
  __syncthreads();
  float sa = aw1b[t], sb = ab1b[t], sc = cw1b[t], sd = cb1b[t];
  for (int k = 0; k < E_DIM; ++k) {
    const float zk = zl[k];
    sa += zk * aw1[k * H_DIM + t];
    sb += zk * ab1[k * H_DIM + t];
    sc += zk * cw1[k * H_DIM + t];
    sd += zk * cb1[k * H_DIM + t];
  }
  h_aT[t * B_N + b] = fmaxf(sa, 0.f);
  hb[t] = fmaxf(sb, 0.f);
  hc[t] = fmaxf(sc, 0.f);
  hcb[t] = fmaxf(sd, 0.f);
  __syncthreads();
  for (int o = t; o < A_DIM; o += H_DIM) {
    float acc = ab2b[o];
    for (int h = 0; h < H_DIM; ++h) acc += hb[h] * ab2[h * A_DIM + o];
    baF[o * B_N + b] += acc;
  }
  float vacc = 0.f;
  for (int i = t; i < S_DIM; i += H_DIM) {
    float w = cw2b[i];
    for (int h = 0; h < H_DIM; ++h) w += hc[h] * cw2[h * S_DIM + i];
    vacc += w * obs[b * S_DIM + i];
  }
  red[t] = vacc;
  __syncthreads();
  for (int s = 64; s > 0; s >>= 1) {
    if (t < s) red[t] += red[t + s];
    __syncthreads();
  }
  if (t == 0) {
    float bv = cb2b[0];
    for (int h = 0; h < H_DIM; ++h) bv += hcb[h] * cb2[h];
    out_v[b] = red[0] + bv;
  }
}

#if USE_TDM
// Issue one 2D TDM tile load: tileH rows x tileW elements (fp32), row stride in elements.
// padInterval/padAmount are the encoded D# codes (interval: 4->32dw, 7->256dw; amount: 3->4dw, 15->16dw).
__device__ __forceinline__ void tdm_load_2d(const void* gptr, unsigned ldsOff,
                                            unsigned tileW, unsigned tileH,
                                            unsigned rowStride,
                                            unsigned padInterval, unsigned padAmount) {
  const unsigned long long ga = (unsigned long long)(uintptr_t)gptr;
  v4u g0;
  g0.x = 1u;                                    // count=1 (valid), user mode
  g0.y = ldsOff;                                // lds_addr (bytes)
  g0.z = (unsigned)ga;                          // global_addr[31:0]
  g0.w = (unsigned)(ga >> 32) | (2u << 30);     // global_addr[56:32] | type=2
  v8i g1;
  g1[0] = (2 << 16) | (1 << 20) |               // data_size=4B, pad_enable
          ((int)padInterval << 22) | ((int)padAmount << 25);
  g1[1] = (int)((tileW & 0xffffu) << 16);       // tensor_dim0[15:0]
  g1[2] = (int)((tileW >> 16) | ((tileH & 0xffffu) << 16)); // dim0 hi | tensor_dim1 lo
  g1[3] = (int)((tileH >> 16) | (tileW << 16)); // dim1 hi | tile_dim0
  g1[4] = (int)tileH;                           // tile_dim1 (tile_dim2=0)
  g1[5] = (int)rowStride;                       // tensor_dim0_stride[31:0]
  g1[6] = 0;
  g1[7] = 0;
  const v4i gz  = {0, 0, 0, 0};
  const v8i gz8 = {0, 0, 0, 0, 0, 0, 0, 0};
  __builtin_amdgcn_tensor_load_to_lds(g0, g1, gz, gz, gz8, 0);
}
#endif

// ---------------- the big one: per-o WMMA GEMM + h_a contraction ----------------
// block o: G_o = aw2[:, o*1024:(o+1)*1024] (128x1024) @ obsT (1024x256)
//          logits[b,o] = sum_h h_aT[h,b]*G_o[h,b] + baF[o,b]
#define ALDS_STRIDE 36                 // even, x4: float4 stores + b64 operand loads, conflict-free
#define BPAIR_STRIDE 544               // 512 + 32: half-waves hit disjoint bank sets
#define BUF_FLOATS (H_DIM * ALDS_STRIDE + 16 * BPAIR_STRIDE)   // 4608 + 8704 = 13312

__global__ void __launch_bounds__(256, 1) k_hyper_gemm(
    const float* __restrict__ aw2, const float* __restrict__ obsP,
    const float* __restrict__ h_aT, const float* __restrict__ baF,
    float* __restrict__ logits) {
#if USE_TDM
  __shared__ float smem[2 * BUF_FLOATS];   // double-buffered: ~104 KB (WGP has 320 KB)
#else
  __shared__ float smem[BUF_FLOATS];
#endif
  float* part = smem;                // alias buf0 A region after main loop (16*256 <= 4608)

  const int o    = blockIdx.x;
  const int t    = threadIdx.x;      // 256 threads = 8 waves
  const int w    = t >> 5;           // wave -> h-strip [16w, 16w+16)
  const int lane = t & 31;
  const int half = lane >> 4;
  const int m    = lane & 15;

  v8f acc[16];
#pragma unroll
  for (int n = 0; n < 16; ++n)
#pragma unroll
    for (int q = 0; q < 8; ++q) acc[n][q] = 0.f;

  const float* aBase = aw2 + (size_t)o * S_DIM;
  const int arow = 16 * w + m;

  // inner compute on one staged chunk: preload all 16 B operands (batched ds loads,
  // single dscnt wait), then an uninterrupted 16-WMMA burst.
#define CHUNK_COMPUTE(aL, bP)                                                     \
  _Pragma("unroll")                                                               \
  for (int kk = 0; kk < 32; kk += 4) {                                            \
    const v2f av = *(const v2f*)&(aL)[arow * ALDS_STRIDE + kk + 2 * half];        \
    const float* bRow = &(bP)[((kk >> 1) + half) * BPAIR_STRIDE + 2 * m];         \
    v2f bvs[16];                                                                  \
    _Pragma("unroll")                                                             \
    for (int nt = 0; nt < 16; ++nt)                                               \
      bvs[nt] = *(const v2f*)&bRow[nt * 32 + ((nt >= 8) ? 16 : 0)];               \
    _Pragma("unroll")                                                             \
    for (int nt = 0; nt < 16; ++nt)                                               \
      acc[nt] = __builtin_amdgcn_wmma_f32_16x16x4_f32(                            \
          false, av, false, bvs[nt], (short)0, acc[nt], false, false);            \
  }

#if USE_TDM
  float* aLds0  = smem;
  float* bPair0 = smem + H_DIM * ALDS_STRIDE;
  float* aLds1  = smem + BUF_FLOATS;
  float* bPair1 = smem + BUF_FLOATS + H_DIM * ALDS_STRIDE;
  const unsigned ldsBase = (unsigned)(uintptr_t)(void*)smem;
  const unsigned aOff0 = ldsBase;
  const unsigned bOff0 = ldsBase + H_DIM * ALDS_STRIDE * 4;
  const unsigned aOff1 = ldsBase + BUF_FLOATS * 4;
  const unsigned bOff1 = ldsBase + (BUF_FLOATS + H_DIM * ALDS_STRIDE) * 4;

#define TDM_ISSUE(kcv, aOff, bOff)                                                \
  do {                                                                            \
    tdm_load_2d(aBase + (kcv), (aOff), 32, H_DIM, AW2_STRIDE, 4, 3);              \
    tdm_load_2d(obsP + (size_t)(kcv) * B_N, (bOff), 512, 16, 512, 7, 15);         \
  } while (0)

  if (w == 0) TDM_ISSUE(0, aOff0, bOff0);
  for (int kc = 0; kc < S_DIM; kc += 64) {
    // ---- half 1: compute buf0(kc), prefetch buf1(kc+32) ----
    if (w == 0) {
      TDM_ISSUE(kc + 32, aOff1, bOff1);            // kc+32 <= 992 always valid
      __builtin_amdgcn_s_wait_tensorcnt(2);        // in-order: buf0's 2 loads done
    }
    __syncthreads();
    CHUNK_COMPUTE(aLds0, bPair0)
    __syncthreads();
    // ---- half 2: compute buf1(kc+32), prefetch buf0(kc+64) ----
    if (w == 0) {
      if (kc + 64 < S_DIM) {
        TDM_ISSUE(kc + 64, aOff0, bOff0);
        __builtin_amdgcn_s_wait_tensorcnt(2);      // buf1's 2 loads done
      } else {
        __builtin_amdgcn_s_wait_tensorcnt(0);
      }
    }
    __syncthreads();
    CHUNK_COMPUTE(aLds1, bPair1)
    __syncthreads();
  }
#else
  float* aLds  = smem;
  float* bPair = smem + H_DIM * ALDS_STRIDE;
  for (int kc = 0; kc < S_DIM; kc += 32) {
    {
      const int row = t >> 1;
      const int seg = (t & 1) * 16;
      const float4* src = (const float4*)(aBase + (size_t)row * AW2_STRIDE + kc + seg);
#pragma unroll
      for (int j = 0; j < 4; ++j)
        *(float4*)&aLds[row * ALDS_STRIDE + seg + 4 * j] = src[j];
    }
    {
      const int kp = t >> 4;                 // pair-row 0..15
      const int q0 = (t & 15) * 32;          // interleaved column chunk
      const float4* src = (const float4*)(obsP + ((size_t)(kc >> 1) + kp) * (2 * B_N) + q0);
      float* dst = &bPair[kp * BPAIR_STRIDE + q0 + ((q0 >= 256) ? 16 : 0)];
#pragma unroll
      for (int j = 0; j < 8; ++j)
        *(float4*)&dst[4 * j] = src[j];
    }
    __syncthreads();
    CHUNK_COMPUTE(aLds, bPair)
    __syncthreads();
  }
#endif

  // C-tile layout: VGPR v, lanes 0-15: M=v, N=lane ; lanes 16-31: M=v+8, N=lane-16
  const int prow = 2 * w + half;
#pragma unroll
  for (int nt = 0; nt < 16; ++nt) {
    const int b = nt * 16 + m;
    float s = 0.f;
#pragma unroll
    for (int v = 0; v < 8; ++v) {
      const int h = 16 * w + 8 * half + v;
      s += h_aT[h * B_N + b] * acc[nt][v];
    }
    part[prow * B_N + b] = s;
  }
  __syncthreads();
  float lg = baF[o * B_N + t];
#pragma unroll
  for (int r = 0; r < 16; ++r) lg += part[r * B_N + t];
  logits[t * A_DIM + o] = lg;
}

// ---------------- log-softmax + gather + entropy ----------------
__global__ void k_softmax(const float* __restrict__ logits, const int* __restrict__ action,
                          float* __restrict__ out_action, float* __restrict__ out_logp,
                          float* __restrict__ out_ent) {
  const int b = blockIdx.x;
  const int t = threadIdx.x;          // 256 threads
  __shared__ float red[256];
  const float* row = logits + b * A_DIM;
  float mx = -3.402823466e38f;
  for (int o = t; o < A_DIM; o += 256) mx = fmaxf(mx, row[o]);
  red[t] = mx; __syncthreads();
  for (int s = 128; s > 0; s >>= 1) { if (t < s) red[t] = fmaxf(red[t], red[t + s]); __syncthreads(); }
  mx = red[0]; __syncthreads();
  float se = 0.f;
  for (int o = t; o < A_DIM; o += 256) se += __expf(row[o] - mx);
  red[t] = se; __syncthreads();
  for (int s = 128; s > 0; s >>= 1) { if (t < s) red[t] += red[t + s]; __syncthreads(); }
  const float lz = mx + __logf(red[0]);
  __syncthreads();
  float ent = 0.f;
  for (int o = t; o < A_DIM; o += 256) { const float lp = row[o] - lz; ent -= __expf(lp) * lp; }
  red[t] = ent; __syncthreads();
  for (int s = 128; s > 0; s >>= 1) { if (t < s) red[t] += red[t + s]; __syncthreads(); }
  if (t == 0) {
    out_ent[b] = red[0];
    const int a = action[b];
    out_logp[b] = row[a] - lz;
    out_action[b] = (float)a;
  }
}

extern "C" void kernel_launch(void* const* d_in, const int* in_sizes, int n_in,
                              void* d_out, int out_size, void* d_ws, size_t ws_size,
                              hipStream_t stream) {
  (void)in_sizes; (void)n_in; (void)out_size; (void)ws_size;
  const float* obs  = (const float*)d_in[0];
  const float* tl   = (const float*)d_in[1];
  const int*   act  = (const int*)  d_in[2];
  const float* we1  = (const float*)d_in[3];  const float* be1  = (const float*)d_in[4];
  const float* we2  = (const float*)d_in[5];  const float* be2  = (const float*)d_in[6];
  const float* aw1  = (const float*)d_in[7];  const float* aw1b = (const float*)d_in[8];
  const float* aw2  = (const float*)d_in[9];  const float* aw2b = (const float*)d_in[10];
  const float* ab1  = (const float*)d_in[11]; const float* ab1b = (const float*)d_in[12];
  const float* ab2  = (const float*)d_in[13]; const float* ab2b = (const float*)d_in[14];
  const float* cw1  = (const float*)d_in[15]; const float* cw1b = (const float*)d_in[16];
  const float* cw2  = (const float*)d_in[17]; const float* cw2b = (const float*)d_in[18];
  const float* cb1  = (const float*)d_in[19]; const float* cb1b = (const float*)d_in[20];
  const float* cb2  = (const float*)d_in[21]; const float* cb2b = (const float*)d_in[22];

  float* ws   = (float*)d_ws;
  float* obsP = ws;                       // 1024*256 (pair-interleaved)
  float* z    = obsP + S_DIM * B_N;       // 256*64
  float* h_aT = z + B_N * E_DIM;          // 128*256
  float* baF  = h_aT + H_DIM * B_N;       // 512*256  (layout [o][b])

  float* out        = (float*)d_out;
  float* out_logits = out;                        // [256,512]
  float* out_action = out + B_N * A_DIM;          // [256]
  float* out_logp   = out_action + B_N;           // [256]
  float* out_ent    = out_logp + B_N;             // [256]
  float* out_v      = out_ent + B_N;              // [256]

  k_transpose<<<dim3(S_DIM / 32, B_N / 32), dim3(32, 8), 0, stream>>>(obs, obsP);
  k_taskmlp<<<B_N, E_DIM, 0, stream>>>(tl, we1, be1, we2, be2, z);
  k_pb<<<A_DIM, B_N, 0, stream>>>(aw2b, obsP, baF);
  k_hidden<<<B_N, H_DIM, 0, stream>>>(z, obs, aw1, aw1b, ab1, ab1b, ab2, ab2b,
                                      cw1, cw1b, cw2, cw2b, cb1, cb1b, cb2, cb2b,
                                      h_aT, baF, out_v);
  k_hyper_gemm<<<A_DIM, 256, 0, stream>>>(aw2, obsP, h_aT, baF, out_logits);
  k_softmax<<<B_N, 256, 0, stream>>>(out_logits, act, out_action, out_logp, out_ent);
}